// GRU_GAT_12008728559867
// MI455X (gfx1250) — compile-verified
//
#include <hip/hip_runtime.h>
#include <hip/hip_bf16.h>

typedef __attribute__((ext_vector_type(2))) float v2f;
typedef __attribute__((ext_vector_type(8))) float v8f;

// ---------------------------------------------------------------------------
// Wave-level fp32 WMMA: D = A(16x4) * B(4x16) + C   (V_WMMA_F32_16X16X4_F32)
// ---------------------------------------------------------------------------
__device__ __forceinline__ v8f wmma_f32_k4(v2f a, v2f b, v8f c) {
  return __builtin_amdgcn_wmma_f32_16x16x4_f32(
      /*neg_a=*/false, a, /*neg_b=*/false, b,
      /*c_mod=*/(short)0, c, /*reuse_a=*/false, /*reuse_b=*/false);
}

// ---------------------------------------------------------------------------
// Generic GEMM: C[M,N] = A[M,K] * W[N,K]^T + bias[N]
// One wave computes a 32x16 slab (two 16x16 M-tiles sharing each B fragment:
// 3 float2 loads feed 2 WMMAs per K-step). Tile order is tm-fastest so the
// weight stream for a given column block is consumed by all M-tiles while it
// is L2-resident -> W streams from HBM exactly once; A (<=4MB) stays in L2.
// Requires M%32==0, N%16==0, K%4==0, lda/ldw even.
// ---------------------------------------------------------------------------
__global__ __launch_bounds__(256)
void gemm_wmma_f32(const float* __restrict__ A, int lda,
                   const float* __restrict__ W, int ldw,
                   const float* __restrict__ bias,
                   float* __restrict__ C, long ldc,
                   int M, int N, int K, int totalWaves)
{
  int wave = blockIdx.x * (blockDim.x >> 5) + (threadIdx.x >> 5);
  if (wave >= totalWaves) return;
  int groupsM = M >> 5;                 // two 16-row tiles per wave
  int gm = wave % groupsM;              // tm-fastest: adjacent waves share W
  int tn = wave / groupsM;
  int lane = threadIdx.x & 31;
  int half = lane >> 4;                 // which K-pair this lane supplies
  int l16  = lane & 15;
  int row0 = gm * 32 + l16;             // A rows (M % 32 == 0)
  int col  = tn * 16 + l16;             // B column (N % 16 == 0)

  const float* Ap0 = A + (size_t)row0 * lda + 2 * half;
  const float* Ap1 = Ap0 + (size_t)16 * lda;
  const float* Wp  = W + (size_t)col * ldw + 2 * half;

  v8f acc0 = {0.f,0.f,0.f,0.f,0.f,0.f,0.f,0.f};
  v8f acc1 = {0.f,0.f,0.f,0.f,0.f,0.f,0.f,0.f};
  for (int k = 0; k < K; k += 4) {
    v2f b  = *(const v2f*)(Wp  + k);
    v2f a0 = *(const v2f*)(Ap0 + k);
    v2f a1 = *(const v2f*)(Ap1 + k);
    acc0 = wmma_f32_k4(a0, b, acc0);
    acc1 = wmma_f32_k4(a1, b, acc1);
  }

  float bv = bias[col];
  // C/D layout: lanes 0-15 hold M=r, lanes 16-31 hold M=8+r, N = l16.
  #pragma unroll
  for (int r = 0; r < 8; ++r) {
    int m = r + 8 * half;
    C[(size_t)(gm * 32 + m)      * ldc + col] = acc0[r] + bv;
    C[(size_t)(gm * 32 + 16 + m) * ldc + col] = acc1[r] + bv;
  }
}

// ---------------------------------------------------------------------------
// GAT node transform: Wx[32768,600] = gather(X, idx) @ Wgat[h,d,f] (as [300,600])
// tn-fastest: the 8 waves of a block share the same 16 gathered X rows (L1).
// ---------------------------------------------------------------------------
__global__ __launch_bounds__(256)
void wx_gemm_wmma(const int* __restrict__ idx, const float* __restrict__ X,
                  const float* __restrict__ Wg, float* __restrict__ Wx,
                  int totalTiles)
{
  const int D = 300, NF = 600, F = 150;
  int wave = blockIdx.x * 8 + (threadIdx.x >> 5);
  if (wave >= totalTiles) return;
  const int tilesN = 38;                 // ceil(600/16): last tile has 8 cols
  int tm = wave / tilesN;
  int tn = wave - tm * tilesN;
  int lane = threadIdx.x & 31;
  int half = lane >> 4, l16 = lane & 15;
  int row = tm * 16 + l16;               // node row, always < 32768
  int col = tn * 16 + l16;
  int colc = col < NF ? col : NF - 1;
  int h = colc / F, f = colc - h * F;    // W_gat[h, d, f] -> B[d, h*F+f]

  const float* Xp = X + (size_t)idx[row] * D + 2 * half;   // node*300+kk even -> aligned
  const float* Wp = Wg + (size_t)h * D * F + (size_t)(2 * half) * F + f;

  v8f acc = {0.f,0.f,0.f,0.f,0.f,0.f,0.f,0.f};
  for (int k = 0; k < D; k += 4) {       // 300/4 = 75 iters, kk+1 <= 299
    v2f a = *(const v2f*)(Xp + k);
    v2f b;
    b.x = Wp[(size_t)k * F];
    b.y = Wp[(size_t)k * F + F];
    acc = wmma_f32_k4(a, b, acc);
  }
  if (col < NF) {
    #pragma unroll
    for (int r = 0; r < 8; ++r) {
      int gm = tm * 16 + r + 8 * half;
      Wx[(size_t)gm * NF + col] = acc[r];
    }
  }
}

// ---------------------------------------------------------------------------
// Per-node attention logits: al_s/al_d[node*4+h] = dot(Wx[node,h,:], a_src/dst[h,:])
// ---------------------------------------------------------------------------
__global__ __launch_bounds__(256)
void al_kernel(const float* __restrict__ Wx, const float* __restrict__ a_src,
               const float* __restrict__ a_dst,
               float* __restrict__ al_s, float* __restrict__ al_d)
{
  int tid = blockIdx.x * blockDim.x + threadIdx.x;
  if (tid >= 32768 * 4) return;
  int node = tid >> 2, h = tid & 3;
  const float* wp = Wx + (size_t)node * 600 + h * 150;
  const float* as = a_src + h * 150;
  const float* ad = a_dst + h * 150;
  float s1 = 0.f, s2 = 0.f;
  for (int f = 0; f < 150; ++f) { float w = wp[f]; s1 += w * as[f]; s2 += w * ad[f]; }
  al_s[tid] = s1; al_d[tid] = s2;
}

// ---------------------------------------------------------------------------
// GAT attention, node 0 only. Reference consumes out[:,0], so only edges with
// dst==0 (plus the node-0 self loop) participate in the softmax/aggregate.
// One block per subgraph g (g = b*S+s), 1024 blocks.
// ---------------------------------------------------------------------------
__global__ __launch_bounds__(256)
void gat0_kernel(const int* __restrict__ edge_index, const float* __restrict__ Wx,
                 const float* __restrict__ al_s, const float* __restrict__ al_d,
                 const float* __restrict__ b_gat, float* __restrict__ gat0)
{
  const int E = 128, N = 32, TE = E + N;
  __shared__ int   s_src[176];
  __shared__ float s_score[4 * 176];
  __shared__ float s_alpha[4 * 176];
  __shared__ int   s_cnt;
  int g = blockIdx.x;
  int t = threadIdx.x;
  if (t == 0) s_cnt = 0;
  __syncthreads();

  if (t < TE) {
    int src, dst;
    if (t < E) { src = edge_index[(size_t)g * 2 * E + t];
                 dst = edge_index[(size_t)g * 2 * E + E + t]; }
    else       { src = dst = t - E; }                     // self loops
    if (dst == 0) {
      int pos = atomicAdd(&s_cnt, 1);
      s_src[pos] = src;
      #pragma unroll
      for (int h = 0; h < 4; ++h) {
        float sc = al_s[(size_t)(g * N + src) * 4 + h] + al_d[(size_t)(g * N) * 4 + h];
        s_score[h * 176 + pos] = sc > 0.f ? sc : 0.2f * sc;   // leaky_relu(0.2)
      }
    }
  }
  __syncthreads();
  int cnt = s_cnt;                       // >= 1 (node-0 self loop)

  if (t < 4) {                           // per-head softmax over cnt entries
    float m = -1e30f;
    for (int i = 0; i < cnt; ++i) m = fmaxf(m, s_score[t * 176 + i]);
    float den = 0.f;
    for (int i = 0; i < cnt; ++i) {
      float e = expf(s_score[t * 176 + i] - m);
      s_alpha[t * 176 + i] = e; den += e;
    }
    float inv = 1.f / den;
    for (int i = 0; i < cnt; ++i) s_alpha[t * 176 + i] *= inv;
  }
  __syncthreads();

  for (int j = t; j < 600; j += blockDim.x) {   // aggregate alpha * Wx[src]
    int h = j / 150;
    float acc = b_gat[j];
    for (int i = 0; i < cnt; ++i)
      acc += s_alpha[h * 176 + i] * Wx[(size_t)(g * N + s_src[i]) * 600 + j];
    gat0[(size_t)g * 600 + j] = acc;
  }
}

// ---------------------------------------------------------------------------
// Build GRU input xs[s*B+b, 0:900] = concat(X[x_indices[b,s,0]], gat0[b*S+s])
// ---------------------------------------------------------------------------
__global__ __launch_bounds__(256)
void signals_kernel(const int* __restrict__ x_indices, const float* __restrict__ X,
                    const float* __restrict__ gat0, float* __restrict__ xs)
{
  int tid = blockIdx.x * blockDim.x + threadIdx.x;
  if (tid >= 1024 * 900) return;
  int row = tid / 900, j = tid - row * 900;     // row = s*32 + b
  int s = row >> 5, b = row & 31;
  int g = b * 32 + s;                            // b*S + s
  float v;
  if (j < 300) v = X[(size_t)x_indices[(size_t)g * 32] * 300 + j];
  else         v = gat0[(size_t)g * 600 + (j - 300)];
  xs[(size_t)row * 900 + j] = v;
}

__global__ __launch_bounds__(256)
void zero_kernel(float* __restrict__ p, int n)
{
  int tid = blockIdx.x * blockDim.x + threadIdx.x;
  if (tid < n) p[tid] = 0.f;
}

// ---------------------------------------------------------------------------
// GRU gate math (PyTorch r,z,n order). In-place h update + write ys[s].
// ---------------------------------------------------------------------------
__global__ __launch_bounds__(256)
void gru_gate_kernel(const float* __restrict__ gi, const float* __restrict__ gh,
                     float* __restrict__ h, float* __restrict__ ys)
{
  int tid = blockIdx.x * blockDim.x + threadIdx.x;
  if (tid >= 32 * 1024) return;
  int b = tid >> 10, u = tid & 1023;
  float i_r = gi[(size_t)b * 3072 + u];
  float i_z = gi[(size_t)b * 3072 + 1024 + u];
  float i_n = gi[(size_t)b * 3072 + 2048 + u];
  float h_r = gh[(size_t)b * 3072 + u];
  float h_z = gh[(size_t)b * 3072 + 1024 + u];
  float h_n = gh[(size_t)b * 3072 + 2048 + u];
  float r = 1.f / (1.f + expf(-(i_r + h_r)));
  float z = 1.f / (1.f + expf(-(i_z + h_z)));
  float n = tanhf(i_n + r * h_n);
  float hp = h[(size_t)b * 1024 + u];
  float hn = (1.f - z) * n + z * hp;
  h[(size_t)b * 1024 + u] = hn;
  ys[(size_t)b * 1024 + u] = hn;
}

// gru_out[b*S+s] = ys1[s*B+b]
__global__ __launch_bounds__(256)
void transpose_kernel(const float* __restrict__ ys, float* __restrict__ gout)
{
  int tid = blockIdx.x * blockDim.x + threadIdx.x;
  if (tid >= 1024 * 1024) return;
  int r = tid >> 10, u = tid & 1023;   // r = b*32 + s
  int b = r >> 5, s = r & 31;
  gout[(size_t)r * 1024 + u] = ys[(size_t)(s * 32 + b) * 1024 + u];
}

// In-place row-wise log_softmax over V=50000; one block per row.
__global__ __launch_bounds__(256)
void log_softmax_kernel(float* __restrict__ out)
{
  const int V = 50000;
  __shared__ float red[256];
  float* p = out + (size_t)blockIdx.x * V;
  int t = threadIdx.x;
  float m = -1e30f;
  for (int j = t; j < V; j += 256) m = fmaxf(m, p[j]);
  red[t] = m; __syncthreads();
  for (int o = 128; o; o >>= 1) { if (t < o) red[t] = fmaxf(red[t], red[t + o]); __syncthreads(); }
  m = red[0]; __syncthreads();
  float s = 0.f;
  for (int j = t; j < V; j += 256) s += expf(p[j] - m);
  red[t] = s; __syncthreads();
  for (int o = 128; o; o >>= 1) { if (t < o) red[t] += red[t + o]; __syncthreads(); }
  float lse = m + logf(red[0]);
  for (int j = t; j < V; j += 256) p[j] = p[j] - lse;
}

// ---------------------------------------------------------------------------
static inline void launch_gemm(const float* A, int lda, const float* W, int ldw,
                               const float* bias, float* C, long ldc,
                               int M, int N, int K, hipStream_t stream)
{
  int waves = (M / 32) * (N / 16);     // one wave per 32x16 slab
  gemm_wmma_f32<<<(waves + 7) / 8, 256, 0, stream>>>(A, lda, W, ldw, bias, C, ldc,
                                                     M, N, K, waves);
}

extern "C" void kernel_launch(void* const* d_in, const int* in_sizes, int n_in,
                              void* d_out, int out_size, void* d_ws, size_t ws_size,
                              hipStream_t stream)
{
  const int*   x_indices  = (const int*)  d_in[0];
  const int*   edge_index = (const int*)  d_in[1];
  const float* X          = (const float*)d_in[2];
  const float* W_gat      = (const float*)d_in[3];
  const float* a_src      = (const float*)d_in[4];
  const float* a_dst      = (const float*)d_in[5];
  const float* b_gat      = (const float*)d_in[6];
  const float* w_ih0      = (const float*)d_in[7];
  const float* w_hh0      = (const float*)d_in[8];
  const float* b_ih0      = (const float*)d_in[9];
  const float* b_hh0      = (const float*)d_in[10];
  const float* w_ih1      = (const float*)d_in[11];
  const float* w_hh1      = (const float*)d_in[12];
  const float* b_ih1      = (const float*)d_in[13];
  const float* b_hh1      = (const float*)d_in[14];
  const float* W_out      = (const float*)d_in[15];
  const float* b_out      = (const float*)d_in[16];
  float* out = (float*)d_out;

  // Workspace layout (floats). The 78.6MB Wx region is dead after
  // signals_kernel, so all GRU buffers alias into it.
  float* ws   = (float*)d_ws;
  float* Wx   = ws;                      // [0, 19,660,800)
  float* gi0  = ws;                      // 3,145,728
  float* gi1  = ws +  3145728;           // 3,145,728
  float* ys0  = ws +  6291456;           // 1,048,576
  float* ys1  = ws +  7340032;           // 1,048,576
  float* hbuf = ws +  8388608;           // 32,768
  float* gh   = ws +  8421376;           // 98,304
  float* gruo = ws +  8519680;           // 1,048,576 (ends 9,568,256 < Wx end)
  float* al_s = ws + 19660800;           // 131,072
  float* al_d = ws + 19791872;           // 131,072
  float* gat0 = ws + 19922944;           // 614,400
  float* xs   = ws + 20537344;           // 921,600 (total 21,458,944 f = 86 MB)

  // 1) GAT node transform (WMMA f32)
  { int tiles = 2048 * 38;
    wx_gemm_wmma<<<(tiles + 7) / 8, 256, 0, stream>>>(x_indices, X, W_gat, Wx, tiles); }
  // 2) attention logits
  al_kernel<<<(32768 * 4 + 255) / 256, 256, 0, stream>>>(Wx, a_src, a_dst, al_s, al_d);
  // 3) node-0 attention + aggregate
  gat0_kernel<<<1024, 256, 0, stream>>>(edge_index, Wx, al_s, al_d, b_gat, gat0);
  // 4) GRU input sequence  [S,B,900]
  signals_kernel<<<(1024 * 900 + 255) / 256, 256, 0, stream>>>(x_indices, X, gat0, xs);

  // 5) GRU layer 0
  launch_gemm(xs, 900, w_ih0, 900, b_ih0, gi0, 3072, 1024, 3072, 900, stream);
  zero_kernel<<<(32768 + 255) / 256, 256, 0, stream>>>(hbuf, 32768);
  for (int s = 0; s < 32; ++s) {
    launch_gemm(hbuf, 1024, w_hh0, 1024, b_hh0, gh, 3072, 32, 3072, 1024, stream);
    gru_gate_kernel<<<(32 * 1024 + 255) / 256, 256, 0, stream>>>(
        gi0 + (size_t)s * 32 * 3072, gh, hbuf, ys0 + (size_t)s * 32 * 1024);
  }

  // 6) GRU layer 1
  launch_gemm(ys0, 1024, w_ih1, 1024, b_ih1, gi1, 3072, 1024, 3072, 1024, stream);
  zero_kernel<<<(32768 + 255) / 256, 256, 0, stream>>>(hbuf, 32768);
  for (int s = 0; s < 32; ++s) {
    launch_gemm(hbuf, 1024, w_hh1, 1024, b_hh1, gh, 3072, 32, 3072, 1024, stream);
    gru_gate_kernel<<<(32 * 1024 + 255) / 256, 256, 0, stream>>>(
        gi1 + (size_t)s * 32 * 3072, gh, hbuf, ys1 + (size_t)s * 32 * 1024);
  }

  // 7) [S,B,U] -> [B*S,U], output GEMM (dominant: streams 205MB W_out), softmax
  transpose_kernel<<<(1024 * 1024 + 255) / 256, 256, 0, stream>>>(ys1, gruo);
  launch_gemm(gruo, 1024, W_out, 1024, b_out, out, 50000, 1024, 50000, 1024, stream);
  log_softmax_kernel<<<1024, 256, 0, stream>>>(out);
}